// ExternalMemory_46059229282411
// MI455X (gfx1250) — compile-verified
//
#include <hip/hip_runtime.h>

// Problem constants (from reference setup)
#define H_    16
#define SEG_  2048
#define D_    128
#define MEM_  8
#define TOT_  (MEM_ * SEG_)            // 16384 positions in cache
#define CPY_  (TOT_ - SEG_)            // 14336 copy-rows per (tensor, head)
#define RPW_  8                        // rows per wave (bulk copy)

// ---------------------------------------------------------------------------
// Kernel 1: bulk cache shift/copy via CDNA5 async global<->LDS data movers.
// Grid: (CPY_/(8*RPW_), H, 2)  -> t = blockIdx.z, h = blockIdx.y, row group
// from blockIdx.x. One wave = 8 consecutive rows (8 * 512B); each lane moves
// 16B per row with global_load_async_to_lds_b128 / store_async_from_lds_b128,
// synchronized with s_wait_asynccnt (ASYNCcnt). No VGPR data path, no div.
// Both cm>=MEM (shift) and cm<MEM (slotted) copy exactly CPY_ rows per (t,h),
// so the grid is cm-independent; cm is read on-device (graph-capture safe).
// ---------------------------------------------------------------------------
__global__ __launch_bounds__(256)
void kv_bulk_copy_async(const float* __restrict__ keys,
                        const float* __restrict__ values,
                        const int*   __restrict__ cur_mem,
                        float*       __restrict__ out)
{
    __shared__ float lds[8 /*waves*/ * RPW_ * D_];   // 32 KB per workgroup

    const int lane = threadIdx.x & 31;
    const int wid  = threadIdx.x >> 5;
    const int t    = blockIdx.z;                       // 0 = keys, 1 = values
    const int h    = blockIdx.y;
    const int j0   = (blockIdx.x * 8 + wid) * RPW_;    // first copy-row index

    const int  cm    = *cur_mem;                       // uniform scalar
    const bool shift = (cm >= MEM_);
    const int  slotB = cm * SEG_;                      // slotted-path boundary

    const float* src_tensor = (t == 0) ? keys : values;

    // LDS byte address of this wave's staging region (low 32 bits of the
    // generic pointer are the LDS offset per the flat-aperture mapping).
    const unsigned ldsBase = (unsigned)(size_t)(&lds[wid * RPW_ * D_]);

    // Stage 1: enqueue 8 async DMA loads (global -> LDS).
    for (int i = 0; i < RPW_; ++i) {
        const int j       = j0 + i;
        const int out_pos = shift ? j : (j < slotB ? j : j + SEG_);
        const int src_pos = shift ? (j + SEG_) : out_pos;
        const float* sp = src_tensor + ((long long)h * TOT_ + src_pos) * D_ + lane * 4;
        const unsigned long long ga = (unsigned long long)(size_t)sp;
        const unsigned ldsOff = ldsBase + (unsigned)((i * D_ + lane * 4) * 4);
        asm volatile("global_load_async_to_lds_b128 %0, %1, off"
                     :: "v"(ldsOff), "v"(ga) : "memory");
    }

    // LDS writes by async loads are not ordered with LDS reads by async
    // stores -> drain ASYNCcnt before turning the buffer around.
    asm volatile("s_wait_asynccnt 0x0" ::: "memory");

    // Stage 2: enqueue 8 async DMA stores (LDS -> global).
    for (int i = 0; i < RPW_; ++i) {
        const int j       = j0 + i;
        const int out_pos = shift ? j : (j < slotB ? j : j + SEG_);
        float* dp = out + (((long long)t * H_ + h) * TOT_ + out_pos) * D_ + lane * 4;
        const unsigned long long ga = (unsigned long long)(size_t)dp;
        const unsigned ldsOff = ldsBase + (unsigned)((i * D_ + lane * 4) * 4);
        asm volatile("global_store_async_from_lds_b128 %0, %1, off"
                     :: "v"(ga), "v"(ldsOff) : "memory");
    }

    // Ensure stores complete before the wave retires.
    asm volatile("s_wait_asynccnt 0x0" ::: "memory");
}

// ---------------------------------------------------------------------------
// Kernel 2: RoPE-rotate the incoming key segment and append it, plus append v.
// Grid: (SEG_/8, H, 2). One wave per output row; lane l owns floats
// [4l, 4l+4). The rotate_half partner x[d +/- 64] lives in lane l^16
// (wave32), fetched with __shfl_xor instead of a second global load.
//   d <  64 : out = x[d]*cos[d] - x[d+64]*sin[d]
//   d >= 64 : out = x[d]*cos[d] + x[d-64]*sin[d]
// ---------------------------------------------------------------------------
__global__ __launch_bounds__(256)
void kv_new_segment(const float* __restrict__ unk,
                    const float* __restrict__ v,
                    const float* __restrict__ cosc,
                    const float* __restrict__ sinc,
                    const int*   __restrict__ pos_ids,
                    const int*   __restrict__ cur_mem,
                    float*       __restrict__ out)
{
    const int lane = threadIdx.x & 31;
    const int wid  = threadIdx.x >> 5;
    const int t    = blockIdx.z;                      // 0 = rope(k), 1 = v
    const int h    = blockIdx.y;
    const int p    = blockIdx.x * 8 + wid;            // position within segment

    const int cm   = *cur_mem;
    const int slot = (cm >= MEM_) ? (MEM_ - 1) : cm;
    const int out_pos = slot * SEG_ + p;
    const int d = lane * 4;

    float4 res;
    if (t == 0) {
        const float* xrow = unk + ((long long)h * SEG_ + p) * D_;
        const int pid = pos_ids[p];
        const float4 x = *(const float4*)(xrow + d);
        float4 xp;                                    // partner half via lanes
        xp.x = __shfl_xor(x.x, 16, 32);
        xp.y = __shfl_xor(x.y, 16, 32);
        xp.z = __shfl_xor(x.z, 16, 32);
        xp.w = __shfl_xor(x.w, 16, 32);
        const float4 c = *(const float4*)(cosc + (long long)pid * D_ + d);
        const float4 s = *(const float4*)(sinc + (long long)pid * D_ + d);
        const float sg = (lane < 16) ? -1.0f : 1.0f;
        res.x = fmaf(sg * xp.x, s.x, x.x * c.x);
        res.y = fmaf(sg * xp.y, s.y, x.y * c.y);
        res.z = fmaf(sg * xp.z, s.z, x.z * c.z);
        res.w = fmaf(sg * xp.w, s.w, x.w * c.w);
    } else {
        res = *(const float4*)(v + ((long long)h * SEG_ + p) * D_ + d);
    }
    *(float4*)(out + (((long long)t * H_ + h) * TOT_ + out_pos) * D_ + d) = res;
}

// ---------------------------------------------------------------------------
// Host-side launcher. Inputs (setup_inputs order):
//   0 keys [1,16,16384,128] f32     1 values [1,16,16384,128] f32
//   2 un_rotated_k [1,16,2048,128]  3 v [1,16,2048,128]
//   4 cos_cache [4096,128]          5 sin_cache [4096,128]
//   6 position_ids [1,2048] int     7 current_memory (scalar int)
// Output: [2,1,16,16384,128] f32
// ---------------------------------------------------------------------------
extern "C" void kernel_launch(void* const* d_in, const int* in_sizes, int n_in,
                              void* d_out, int out_size, void* d_ws, size_t ws_size,
                              hipStream_t stream)
{
    (void)in_sizes; (void)n_in; (void)out_size; (void)d_ws; (void)ws_size;

    const float* keys   = (const float*)d_in[0];
    const float* values = (const float*)d_in[1];
    const float* unk    = (const float*)d_in[2];
    const float* v      = (const float*)d_in[3];
    const float* cosc   = (const float*)d_in[4];
    const float* sinc   = (const float*)d_in[5];
    const int*   pid    = (const int*)d_in[6];
    const int*   cm     = (const int*)d_in[7];
    float* out = (float*)d_out;

    // Bulk copy: per (t,h) CPY_=14336 rows, 64 rows/block -> 224 x 16 x 2.
    kv_bulk_copy_async<<<dim3(CPY_ / (8 * RPW_), H_, 2), 256, 0, stream>>>(
        keys, values, cm, out);

    // New segment: per (t,h) SEG_=2048 rows, 8 rows/block -> 256 x 16 x 2.
    kv_new_segment<<<dim3(SEG_ / 8, H_, 2), 256, 0, stream>>>(
        unk, v, cosc, sinc, pid, cm, out);
}